// KGATLayer_42966852829692
// MI455X (gfx1250) — compile-verified
//
#include <hip/hip_runtime.h>
#include <math.h>

// ---------------------------------------------------------------------------
// KGAT layer for MI455X (gfx1250, wave32).
// Algebraically folded: all per-edge dense math reduced to per-node /
// per-relation precomputes done with V_WMMA_F32_16X16X4_F32 (fp32-exact),
// edge passes are pure gather + L2-resident f32 atomics.
// ---------------------------------------------------------------------------

#define NUM_NODES 100000
#define NUM_RELS  500
#define DIM       64
#define NUM_EDGES 1000000

typedef __attribute__((ext_vector_type(2))) float v2f;
typedef __attribute__((ext_vector_type(8))) float v8f;

// ---------------- zero fill ----------------
__global__ void zero_kernel(float* __restrict__ p, int n) {
  for (int i = blockIdx.x * blockDim.x + threadIdx.x; i < n;
       i += gridDim.x * blockDim.x)
    p[i] = 0.0f;
}

// ---------------- weight folding (single block) ----------------
// M_s = Wfc_s @ W_ent      (64x64)   u_s = ent_embed @ M_s^T
// M_d = Wfc_d @ W_ent      (64x64)
// M_r = Wfc_r @ W_relL     (64x64)
// M_2 = W_rel2 @ W_relL    (64x64)   rel2_proj = rel_embed @ M_2^T + v_rel2
// cvec = b_fc + Wfc_s@b_ent + Wfc_d@b_ent + Wfc_r@b_relL
// v_rel2 = W_rel2@b_relL + b_rel2
// a0 = cvec . w_a + b_a
__global__ __launch_bounds__(256) void fold_kernel(
    const float* __restrict__ W_ent, const float* __restrict__ b_ent,
    const float* __restrict__ W_relL, const float* __restrict__ b_relL,
    const float* __restrict__ W_rel2, const float* __restrict__ b_rel2,
    const float* __restrict__ W_a, const float* __restrict__ b_a,
    const float* __restrict__ W_fc, const float* __restrict__ b_fc,
    float* __restrict__ M_s, float* __restrict__ M_d,
    float* __restrict__ M_r, float* __restrict__ M_2,
    float* __restrict__ cvec, float* __restrict__ v_rel2,
    float* __restrict__ a0) {
  __shared__ float scvec[DIM];
  const int t = threadIdx.x;
  for (int idx = t; idx < DIM * DIM; idx += 256) {
    const int o = idx >> 6, k = idx & 63;
    float ms = 0.f, md = 0.f, mr = 0.f, m2 = 0.f;
    for (int u = 0; u < DIM; ++u) {
      const float we = W_ent[u * DIM + k];
      const float wl = W_relL[u * DIM + k];
      ms += W_fc[o * 192 + u] * we;
      md += W_fc[o * 192 + 64 + u] * we;
      mr += W_fc[o * 192 + 128 + u] * wl;
      m2 += W_rel2[o * DIM + u] * wl;
    }
    M_s[idx] = ms; M_d[idx] = md; M_r[idx] = mr; M_2[idx] = m2;
  }
  if (t < DIM) {
    float cv = b_fc[t], v2 = b_rel2[t];
    for (int u = 0; u < DIM; ++u) {
      cv += W_fc[t * 192 + u] * b_ent[u]
          + W_fc[t * 192 + 64 + u] * b_ent[u]
          + W_fc[t * 192 + 128 + u] * b_relL[u];
      v2 += W_rel2[t * DIM + u] * b_relL[u];
    }
    cvec[t] = cv; v_rel2[t] = v2; scvec[t] = cv;
  }
  __syncthreads();
  if (t == 0) {
    float a = b_a[0];
    for (int o = 0; o < DIM; ++o) a += scvec[o] * W_a[o];
    a0[0] = a;
  }
}

// ---------------- WMMA GEMM: out[M,64] = A[M,64] @ W[64,64]^T (+bias) -------
// One wave computes a 16x64 output tile via 4 accumulators of
// V_WMMA_F32_16X16X4_F32, K swept in steps of 4 (16 iters).
// A fragment (16x4 f32, ISA 7.12.2): lanes 0-15 carry K={k,k+1},
// lanes 16-31 carry K={k+2,k+3}, M = lane&15.
// B fragment (4x16): v0 = rows K/K+2 striped over lane halves, v1 = K+1/K+3,
// N = lane&15; B[k][n] = W[n][k], staged through LDS (16 KB).
__global__ __launch_bounds__(256) void wmma_gemm64(
    const float* __restrict__ A, const float* __restrict__ W,
    const float* __restrict__ bias, float* __restrict__ out, int M) {
  __shared__ float sW[DIM * DIM];
  for (int i = threadIdx.x; i < DIM * DIM; i += 256) sW[i] = W[i];
  __syncthreads();

  const int wave = threadIdx.x >> 5;
  const int lane = threadIdx.x & 31;
  const int m0 = (blockIdx.x * 8 + wave) * 16;
  if (m0 >= M) return;  // wave-uniform: EXEC stays all-ones for WMMA

  const int lm = lane & 15;
  const int hi = lane >> 4;  // 0: low lane half, 1: high lane half

  int arow = m0 + lm;
  if (arow >= M) arow = M - 1;  // clamp (stores are guarded below)
  const float* __restrict__ Arow = A + (size_t)arow * DIM;

  v8f acc0 = {}, acc1 = {}, acc2 = {}, acc3 = {};
  for (int k = 0; k < DIM; k += 4) {
    v2f a;
    a.x = Arow[k + 2 * hi + 0];
    a.y = Arow[k + 2 * hi + 1];
    v2f b0, b1, b2, b3;
    const float* w0 = &sW[(0 * 16 + lm) * DIM + k + 2 * hi];
    const float* w1 = &sW[(1 * 16 + lm) * DIM + k + 2 * hi];
    const float* w2 = &sW[(2 * 16 + lm) * DIM + k + 2 * hi];
    const float* w3 = &sW[(3 * 16 + lm) * DIM + k + 2 * hi];
    b0.x = w0[0]; b0.y = w0[1];
    b1.x = w1[0]; b1.y = w1[1];
    b2.x = w2[0]; b2.y = w2[1];
    b3.x = w3[0]; b3.y = w3[1];
    acc0 = __builtin_amdgcn_wmma_f32_16x16x4_f32(false, a, false, b0,
                                                 (short)0, acc0, false, false);
    acc1 = __builtin_amdgcn_wmma_f32_16x16x4_f32(false, a, false, b1,
                                                 (short)0, acc1, false, false);
    acc2 = __builtin_amdgcn_wmma_f32_16x16x4_f32(false, a, false, b2,
                                                 (short)0, acc2, false, false);
    acc3 = __builtin_amdgcn_wmma_f32_16x16x4_f32(false, a, false, b3,
                                                 (short)0, acc3, false, false);
  }

  v8f accs[4] = {acc0, acc1, acc2, acc3};
#pragma unroll
  for (int ob = 0; ob < 4; ++ob) {
    const float bv = bias ? bias[ob * 16 + lm] : 0.0f;
#pragma unroll
    for (int v = 0; v < 8; ++v) {
      const int row = m0 + v + 8 * hi;  // C/D layout: VGPR v -> rows v, v+8
      if (row < M)
        out[(size_t)row * DIM + ob * 16 + lm] = accs[ob][v] + bv;
    }
  }
}

// ---------------- per-row dot with 64-vector ----------------
__global__ void dot64_kernel(const float* __restrict__ X,
                             const float* __restrict__ v,
                             float* __restrict__ out, int M) {
  const int m = blockIdx.x * blockDim.x + threadIdx.x;
  if (m >= M) return;
  const float* x = X + (size_t)m * DIM;
  float acc = 0.f;
#pragma unroll 8
  for (int j = 0; j < DIM; ++j) acc += x[j] * v[j];
  out[m] = acc;
}

// ---------------- edge pass 1: logits, exp, b_sum / counts ----------------
__global__ void edge_logits_kernel(
    const int* __restrict__ src, const int* __restrict__ dst,
    const int* __restrict__ rel, const float* __restrict__ a_s,
    const float* __restrict__ a_d, const float* __restrict__ a_r,
    const float* __restrict__ a0p, float* __restrict__ b_e,
    float* __restrict__ b_sum, float* __restrict__ cnt, int E) {
  const float a0 = a0p[0];
  for (int e = blockIdx.x * blockDim.x + threadIdx.x; e < E;
       e += gridDim.x * blockDim.x) {
    const int s = src[e], d = dst[e], r = rel[e];
    float l = a_s[s] + a_d[d] + a_r[r] + a0;
    l = l > 0.f ? l : 0.01f * l;  // leaky_relu(0.01)
    const float b = expf(l);
    b_e[e] = b;
    atomicAdd(&b_sum[s], b);
    atomicAdd(&cnt[r], 1.0f);
  }
}

// ---------------- edge pass 2: h_ent[s] += alpha * c ----------------
// One wave per edge iteration: lanes cover the 64 dims in two coalesced halves.
__global__ __launch_bounds__(256) void edge_scatter_h_kernel(
    const int* __restrict__ src, const int* __restrict__ dst,
    const int* __restrict__ rel, const float* __restrict__ u_s,
    const float* __restrict__ u_d, const float* __restrict__ u_r,
    const float* __restrict__ cvec, const float* __restrict__ b_e,
    const float* __restrict__ b_sum, float* __restrict__ h_ent, int E) {
  const int gw = blockIdx.x * (blockDim.x >> 5) + (threadIdx.x >> 5);
  const int lane = threadIdx.x & 31;
  const int nw = gridDim.x * (blockDim.x >> 5);
  for (int e = gw; e < E; e += nw) {
    const int s = src[e], d = dst[e], r = rel[e];
    const float alpha = b_e[e] / b_sum[s];
    const float* ps = u_s + (size_t)s * DIM;
    const float* pd = u_d + (size_t)d * DIM;
    const float* pr = u_r + (size_t)r * DIM;
    float* ph = h_ent + (size_t)s * DIM;
#pragma unroll
    for (int half = 0; half < 2; ++half) {
      const int j = lane + 32 * half;
      const float c = ps[j] + pd[j] + pr[j] + cvec[j];
      atomicAdd(&ph[j], alpha * c);
    }
  }
}

// ---------------- edge pass 3: relation feature sums ----------------
__global__ __launch_bounds__(256) void edge_featsum_kernel(
    const int* __restrict__ src, const int* __restrict__ dst,
    const int* __restrict__ rel, const float* __restrict__ h_ent,
    float* __restrict__ featsum, int E) {
  const int gw = blockIdx.x * (blockDim.x >> 5) + (threadIdx.x >> 5);
  const int lane = threadIdx.x & 31;
  const int nw = gridDim.x * (blockDim.x >> 5);
  for (int e = gw; e < E; e += nw) {
    const int s = src[e], d = dst[e], r = rel[e];
    const float* hs = h_ent + (size_t)s * DIM;
    const float* hd = h_ent + (size_t)d * DIM;
    float* fs = featsum + (size_t)r * 128;
#pragma unroll
    for (int half = 0; half < 2; ++half) {
      const int j = lane + 32 * half;
      atomicAdd(&fs[j], hs[j]);
      atomicAdd(&fs[64 + j], hd[j]);
    }
  }
}

// ---------------- h_rel = mean-feats @ W_rel3^T + b_rel3 ----------------
// rel2 third of the 192-wide mean is rel2_proj[r] (constant per relation),
// zero when the relation has no edges (reference: mean = sums/max(cnt,1)).
__global__ void h_rel_kernel(const float* __restrict__ featsum,
                             const float* __restrict__ cnt,
                             const float* __restrict__ rel2_proj,
                             const float* __restrict__ W_rel3,
                             const float* __restrict__ b_rel3,
                             float* __restrict__ h_rel, int R) {
  const int t = blockIdx.x * blockDim.x + threadIdx.x;
  if (t >= R * DIM) return;
  const int r = t >> 6, o = t & 63;
  const float c = cnt[r];
  const float inv = 1.0f / fmaxf(c, 1.0f);
  const float* fs = featsum + (size_t)r * 128;
  const float* w = W_rel3 + (size_t)o * 192;
  float acc = b_rel3[o];
  for (int j = 0; j < 128; ++j) acc += (fs[j] * inv) * w[j];
  if (c > 0.f) {
    const float* r2 = rel2_proj + (size_t)r * DIM;
    for (int j = 0; j < DIM; ++j) acc += r2[j] * w[128 + j];
  }
  h_rel[t] = acc;
}

// ---------------- workspace layout (floats) ----------------
#define OFF_MS     0
#define OFF_MD     4096
#define OFF_MR     8192
#define OFF_M2     12288
#define OFF_CVEC   16384
#define OFF_VREL2  16448
#define OFF_A0     16512
#define OFF_UR     16576                 // 500*64
#define OFF_AR     48576                 // 500 (+pad)
#define OFF_REL2P  49088                 // 500*64
#define OFF_AS     81088                 // 100000
#define OFF_AD     181088                // 100000
#define OFF_BSUM   281088                // 100000   -- zeroed region start
#define OFF_CNT    381088                // 500
#define OFF_FEAT   381588                // 500*128  -- zeroed region end
#define OFF_BE     445588                // 1000000
#define OFF_US     1445588               // 100000*64
#define OFF_UD     7845588               // 100000*64  (end 14245588 fl ~ 57MB)
#define ZERO_CNT   (OFF_FEAT + NUM_RELS * 128 - OFF_BSUM)

extern "C" void kernel_launch(void* const* d_in, const int* in_sizes, int n_in,
                              void* d_out, int out_size, void* d_ws,
                              size_t ws_size, hipStream_t stream) {
  const float* ent_embed = (const float*)d_in[0];
  const float* rel_embed = (const float*)d_in[1];
  const float* W_ent  = (const float*)d_in[2];
  const float* b_ent  = (const float*)d_in[3];
  const float* W_relL = (const float*)d_in[4];
  const float* b_relL = (const float*)d_in[5];
  const float* W_rel2 = (const float*)d_in[6];
  const float* b_rel2 = (const float*)d_in[7];
  const float* W_rel3 = (const float*)d_in[8];
  const float* b_rel3 = (const float*)d_in[9];
  const float* W_a    = (const float*)d_in[10];
  const float* b_a    = (const float*)d_in[11];
  const float* W_fc   = (const float*)d_in[12];
  const float* b_fc   = (const float*)d_in[13];
  const int* src_idx  = (const int*)d_in[14];
  const int* dst_idx  = (const int*)d_in[15];
  const int* rel_idx  = (const int*)d_in[16];

  float* ws = (float*)d_ws;
  float* h_ent = (float*)d_out;                         // [N,64]
  float* h_rel = (float*)d_out + (size_t)NUM_NODES * DIM;  // [R,64]

  // 1) zero accumulators (d_out is poisoned by harness; atomics need 0 base)
  zero_kernel<<<2048, 256, 0, stream>>>(h_ent, NUM_NODES * DIM);
  zero_kernel<<<512, 256, 0, stream>>>(ws + OFF_BSUM, ZERO_CNT);

  // 2) fold weights
  fold_kernel<<<1, 256, 0, stream>>>(
      W_ent, b_ent, W_relL, b_relL, W_rel2, b_rel2, W_a, b_a, W_fc, b_fc,
      ws + OFF_MS, ws + OFF_MD, ws + OFF_MR, ws + OFF_M2, ws + OFF_CVEC,
      ws + OFF_VREL2, ws + OFF_A0);

  // 3) node / relation precomputes on the WMMA pipe
  const int gemm_blocks_N = (NUM_NODES + 127) / 128;  // 8 waves x 16 rows
  const int gemm_blocks_R = (NUM_RELS + 127) / 128;
  wmma_gemm64<<<gemm_blocks_N, 256, 0, stream>>>(ent_embed, ws + OFF_MS,
                                                 nullptr, ws + OFF_US, NUM_NODES);
  wmma_gemm64<<<gemm_blocks_N, 256, 0, stream>>>(ent_embed, ws + OFF_MD,
                                                 nullptr, ws + OFF_UD, NUM_NODES);
  wmma_gemm64<<<gemm_blocks_R, 256, 0, stream>>>(rel_embed, ws + OFF_MR,
                                                 nullptr, ws + OFF_UR, NUM_RELS);
  wmma_gemm64<<<gemm_blocks_R, 256, 0, stream>>>(rel_embed, ws + OFF_M2,
                                                 ws + OFF_VREL2, ws + OFF_REL2P,
                                                 NUM_RELS);

  // 4) attention scalars
  dot64_kernel<<<(NUM_NODES + 255) / 256, 256, 0, stream>>>(ws + OFF_US, W_a,
                                                            ws + OFF_AS, NUM_NODES);
  dot64_kernel<<<(NUM_NODES + 255) / 256, 256, 0, stream>>>(ws + OFF_UD, W_a,
                                                            ws + OFF_AD, NUM_NODES);
  dot64_kernel<<<(NUM_RELS + 255) / 256, 256, 0, stream>>>(ws + OFF_UR, W_a,
                                                           ws + OFF_AR, NUM_RELS);

  // 5-7) edge passes
  edge_logits_kernel<<<4096, 256, 0, stream>>>(
      src_idx, dst_idx, rel_idx, ws + OFF_AS, ws + OFF_AD, ws + OFF_AR,
      ws + OFF_A0, ws + OFF_BE, ws + OFF_BSUM, ws + OFF_CNT, NUM_EDGES);
  edge_scatter_h_kernel<<<4096, 256, 0, stream>>>(
      src_idx, dst_idx, rel_idx, ws + OFF_US, ws + OFF_UD, ws + OFF_UR,
      ws + OFF_CVEC, ws + OFF_BE, ws + OFF_BSUM, h_ent, NUM_EDGES);
  edge_featsum_kernel<<<4096, 256, 0, stream>>>(
      src_idx, dst_idx, rel_idx, h_ent, ws + OFF_FEAT, NUM_EDGES);

  // 8) h_rel
  h_rel_kernel<<<(NUM_RELS * DIM + 255) / 256, 256, 0, stream>>>(
      ws + OFF_FEAT, ws + OFF_CNT, ws + OFF_REL2P, W_rel3, b_rel3, h_rel,
      NUM_RELS);
}